// DepthSeparableConv2d_conv4_2_37572373905424
// MI455X (gfx1250) — compile-verified
//
#include <hip/hip_runtime.h>
#include <math.h>

// ---------------------------------------------------------------------------
// DepthSeparableConv2d on gfx1250 (MI455X):
//   stage 1: depthwise 3x3 + bias + BN + ReLU + plane prune (4.0)
//            -> bandwidth-bound, VALU with LDS halo tile
//   stage 2: pointwise 1x1 = per-batch GEMM C[512x784] = W[512x512]*H[512x784]
//            -> V_WMMA_F32_16X16X4_F32, M-tile = 32 so each B fragment
//               (L2 traffic) feeds two WMMAs (2x FLOP per L2 byte vs M=16)
// ---------------------------------------------------------------------------

#define CIN   512
#define COUT  512
#define IH    28
#define IW    28
#define NPIX  (IH * IW)        // 784
#define NB    64
#define EPSBN 1e-5f

typedef __attribute__((ext_vector_type(2))) float v2f;
typedef __attribute__((ext_vector_type(8))) float v8f;

// ===========================================================================
// Kernel 1: depthwise 3x3 conv + bias + BN1 + ReLU + prune(4.0)
// one 256-thread block per (b, c) plane
// ===========================================================================
__global__ __launch_bounds__(256)
void dw_bn_relu_prune_kernel(const float* __restrict__ x,
                             const float* __restrict__ dw_w,
                             const float* __restrict__ dw_b,
                             const float* __restrict__ g1,
                             const float* __restrict__ b1,
                             const float* __restrict__ m1,
                             const float* __restrict__ v1,
                             float* __restrict__ h_out)
{
    const int plane = blockIdx.x;          // b*CIN + c
    const int c     = plane & (CIN - 1);
    const int tid   = threadIdx.x;         // 0..255

    __shared__ float tile[30 * 30];        // halo'd input plane (zero padded)
    __shared__ float wsm[9];
    __shared__ float red[256];

    const float* xp = x + (size_t)plane * NPIX;

    for (int i = tid; i < 900; i += 256) {
        const int yy = i / 30 - 1;
        const int xx = i % 30 - 1;
        float val = 0.0f;
        if (yy >= 0 && yy < IH && xx >= 0 && xx < IW)
            val = xp[yy * IW + xx];
        tile[i] = val;
    }
    if (tid < 9) wsm[tid] = dw_w[c * 9 + tid];
    __syncthreads();

    const float inv  = g1[c] * rsqrtf(v1[c] + EPSBN);
    const float beta = b1[c] - m1[c] * inv;
    const float bias = dw_b[c];

    float vals[4];
    float amax = 0.0f;                     // post-ReLU values are >= 0
#pragma unroll
    for (int it = 0; it < 4; ++it) {
        const int i = tid + it * 256;
        float r = 0.0f;
        if (i < NPIX) {
            const int y  = i / IW;
            const int xx = i % IW;
            float acc = bias;
#pragma unroll
            for (int dy = 0; dy < 3; ++dy)
#pragma unroll
                for (int dx = 0; dx < 3; ++dx)
                    acc = fmaf(tile[(y + dy) * 30 + (xx + dx)], wsm[dy * 3 + dx], acc);
            r = fmaxf(acc * inv + beta, 0.0f);   // BN + ReLU
        }
        vals[it] = r;
        amax = fmaxf(amax, r);
    }

    red[tid] = amax;
    __syncthreads();
    for (int s = 128; s > 0; s >>= 1) {
        if (tid < s) red[tid] = fmaxf(red[tid], red[tid + s]);
        __syncthreads();
    }
    const float keep = (red[0] < 4.0f) ? 0.0f : 1.0f;

    float* hp = h_out + (size_t)plane * NPIX;
#pragma unroll
    for (int it = 0; it < 4; ++it) {
        const int i = tid + it * 256;
        if (i < NPIX) hp[i] = vals[it] * keep;
    }
}

// ===========================================================================
// Kernel 2: pointwise GEMM (WMMA f32 16x16x4, M-tile 32) + BN2 + ReLU + prune
// grid = (COUT/32, NB); block = 256 threads = 8 waves
// block owns output rows co0..co0+31 for one batch -> plane prune is local
// ===========================================================================
#define MROWS   32
#define WSTRIDE 516   // 512 + 4 pad: A-frag gathers hit 16 distinct LDS banks

__global__ __launch_bounds__(256)
void pw_wmma_bn_relu_prune_kernel(const float* __restrict__ h,
                                  const float* __restrict__ pw_w,
                                  const float* __restrict__ pw_b,
                                  const float* __restrict__ g2,
                                  const float* __restrict__ b2,
                                  const float* __restrict__ m2,
                                  const float* __restrict__ v2,
                                  float* __restrict__ out)
{
    const int co0 = blockIdx.x * MROWS;    // output-channel tile base
    const int b   = blockIdx.y;            // batch
    const int tid = threadIdx.x;
    const int wave = tid >> 5;             // 0..7
    const int lane = tid & 31;
    const int nsub = lane & 15;            // N (pixel) / M (row) sub-index
    const int half = lane >> 4;            // 0: K={0,1}, 1: K={2,3}
    const int koff = half * 2;

    __shared__ float wlds[MROWS * WSTRIDE];   // weight tile [co][ci]  (~66 KB)
    __shared__ float outs[MROWS * NPIX];      // result tile [co][pix] (~100 KB)
    __shared__ float bias2[MROWS], scale2[MROWS], beta2[MROWS];
    __shared__ float red2[256];

    // stage weight tile: W[co0+row][ci], rows 0..31
    for (int i = tid; i < MROWS * CIN; i += 256) {
        const int row = i >> 9;            // /512
        const int ci  = i & (CIN - 1);
        wlds[row * WSTRIDE + ci] = pw_w[(size_t)(co0 + row) * CIN + ci];
    }
    if (tid < MROWS) {
        const int co = co0 + tid;
        const float iv = g2[co] * rsqrtf(v2[co] + EPSBN);
        scale2[tid] = iv;
        beta2[tid]  = b2[co] - m2[co] * iv;
        bias2[tid]  = pw_b[co];
    }
    __syncthreads();

    const float* hb = h + (size_t)b * (CIN * NPIX);

    // 49 N-tiles of 16 pixels, round-robin over 8 waves;
    // each B fragment feeds TWO wmmas (rows 0..15 and 16..31)
    for (int t = wave; t < NPIX / 16; t += 8) {
        const int p0 = t * 16;
        v8f accA = {0.f, 0.f, 0.f, 0.f, 0.f, 0.f, 0.f, 0.f};
        v8f accB = {0.f, 0.f, 0.f, 0.f, 0.f, 0.f, 0.f, 0.f};

        for (int k0 = 0; k0 < CIN; k0 += 4) {
            // A fragments from LDS: lane holds M=nsub, K = koff{+0,+1}
            const float* wpA = &wlds[nsub * WSTRIDE + k0 + koff];
            const float* wpB = &wlds[(16 + nsub) * WSTRIDE + k0 + koff];
            v2f afragA, afragB;
            afragA.x = wpA[0];  afragA.y = wpA[1];
            afragB.x = wpB[0];  afragB.y = wpB[1];
            // B fragment from global (L2): lane holds N=nsub, K = koff{+0,+1}
            const float* hp = hb + (size_t)(k0 + koff) * NPIX + p0 + nsub;
            v2f bfrag;
            bfrag.x = hp[0];
            bfrag.y = hp[NPIX];

            accA = __builtin_amdgcn_wmma_f32_16x16x4_f32(
                false, afragA, false, bfrag, (short)0, accA, false, false);
            accB = __builtin_amdgcn_wmma_f32_16x16x4_f32(
                false, afragB, false, bfrag, (short)0, accB, false, false);
        }

        // epilogue -> LDS result tile (bias + BN2 + ReLU), both row halves
#pragma unroll
        for (int r = 0; r < 8; ++r) {
            const int row = r + half * 8;  // D VGPR r: M = r (lanes 0-15) / r+8
            float vA = (accA[r] + bias2[row]) * scale2[row] + beta2[row];
            float vB = (accB[r] + bias2[16 + row]) * scale2[16 + row] + beta2[16 + row];
            outs[row * NPIX + p0 + nsub]        = fmaxf(vA, 0.0f);
            outs[(16 + row) * NPIX + p0 + nsub] = fmaxf(vB, 0.0f);
        }
    }
    __syncthreads();

    // per-row (per output channel) plane max: 8 threads per row, 32 rows
    {
        const int row  = tid >> 3;         // 0..31
        const int slot = tid & 7;
        float mx = 0.0f;                   // values are post-ReLU (>= 0)
        for (int i = slot; i < NPIX; i += 8)
            mx = fmaxf(mx, outs[row * NPIX + i]);
        red2[tid] = mx;
        __syncthreads();
        for (int s = 4; s > 0; s >>= 1) {
            if (slot < s) red2[tid] = fmaxf(red2[tid], red2[tid + s]);
            __syncthreads();
        }
    }

    // prune + store
    float* op = out + ((size_t)b * COUT + co0) * NPIX;
    for (int i = tid; i < MROWS * NPIX; i += 256) {
        const int row = i / NPIX;
        const float keep = (red2[row * 8] < 0.001f) ? 0.0f : 1.0f;
        op[i] = outs[i] * keep;
    }
}

// ===========================================================================
// launch
// ===========================================================================
extern "C" void kernel_launch(void* const* d_in, const int* in_sizes, int n_in,
                              void* d_out, int out_size, void* d_ws, size_t ws_size,
                              hipStream_t stream)
{
    const float* x    = (const float*)d_in[0];
    const float* dw_w = (const float*)d_in[1];
    const float* dw_b = (const float*)d_in[2];
    const float* g1   = (const float*)d_in[3];
    const float* b1   = (const float*)d_in[4];
    const float* m1   = (const float*)d_in[5];
    const float* v1   = (const float*)d_in[6];
    const float* pw_w = (const float*)d_in[7];
    const float* pw_b = (const float*)d_in[8];
    const float* g2   = (const float*)d_in[9];
    const float* b2   = (const float*)d_in[10];
    const float* m2   = (const float*)d_in[11];
    const float* v2   = (const float*)d_in[12];

    float* out = (float*)d_out;
    float* h   = (float*)d_ws;   // intermediate: 64*512*784 floats (~103 MB)

    dw_bn_relu_prune_kernel<<<NB * CIN, 256, 0, stream>>>(
        x, dw_w, dw_b, g1, b1, m1, v1, h);

    dim3 grid2(COUT / MROWS, NB);
    pw_wmma_bn_relu_prune_kernel<<<grid2, 256, 0, stream>>>(
        h, pw_w, pw_b, g2, b2, m2, v2, out);
}